// HashEmbedder_11768210391672
// MI455X (gfx1250) — compile-verified
//
#include <hip/hip_runtime.h>
#include <stdint.h>
#include <stddef.h>

#define NLV     16
#define NCACHED 6          // levels 0..5 staged in LDS: 54,216 bytes
#define BLOCK   256        // 8 wave32 waves per block
#define NBLOCKS 1024

typedef float v2f __attribute__((ext_vector_type(2)));
typedef float v4f __attribute__((ext_vector_type(4)));
typedef int   v2i __attribute__((ext_vector_type(2)));

typedef __attribute__((address_space(1))) v2i* gptr_v2i;  // global
typedef __attribute__((address_space(3))) v2i* lptr_v2i;  // LDS

#if defined(__has_builtin)
#if __has_builtin(__builtin_amdgcn_global_load_async_to_lds_b64)
#define HAVE_ASYNC_LDS 1
#endif
#if __has_builtin(__builtin_amdgcn_s_wait_asynccnt)
#define HAVE_WAIT_ASYNCCNT 1
#endif
#endif

// res_i = floor(16 * (2^(1/3))^i)  (matches the reference's float32 math)
constexpr int kRes[NLV] = {16, 20, 25, 32, 40, 50, 64, 80,
                           101, 128, 161, 203, 256, 322, 406, 512};
// (res+1)^2 entries for the LDS-cached levels, and their offsets (in float2)
constexpr int kEntries[NCACHED] = {289, 441, 676, 1089, 1681, 2601};
constexpr int kOff[NCACHED]     = {0, 289, 730, 1406, 2495, 4176};
#define LDS_TOTAL 6777     // sum of kEntries

struct Tabs { const v2f* t[NLV]; };

__global__ __launch_bounds__(BLOCK) void hashgrid_kernel(
    const v2f* __restrict__ x, Tabs tabs, float* __restrict__ out, int npts)
{
    __shared__ v2f lds_tab[LDS_TOTAL];

    // ---- Stage the 6 smallest tables into LDS (CDNA5 async copy path) ----
    #pragma unroll
    for (int l = 0; l < NCACHED; ++l) {
        const v2f* __restrict__ src = tabs.t[l];
        v2f* dst = &lds_tab[kOff[l]];
        const int n = kEntries[l];
#ifdef HAVE_ASYNC_LDS
        for (int j = threadIdx.x; j < n; j += BLOCK) {
            __builtin_amdgcn_global_load_async_to_lds_b64(
                (gptr_v2i)(src + j),
                (lptr_v2i)(dst + j),
                /*offset=*/0, /*cpol=*/0);
        }
#else
        for (int j = threadIdx.x; j < n; j += BLOCK) dst[j] = src[j];
#endif
    }
#ifdef HAVE_ASYNC_LDS
#ifdef HAVE_WAIT_ASYNCCNT
    __builtin_amdgcn_s_wait_asynccnt(0);
#else
    asm volatile("s_wait_asynccnt 0" ::: "memory");
#endif
#endif
    __syncthreads();

    // ---- Grid-stride over points ----
    const int stride = (int)gridDim.x * BLOCK;
    for (int p = (int)blockIdx.x * BLOCK + (int)threadIdx.x; p < npts; p += stride) {
        // streaming input: non-temporal, don't pollute L2 (tables live there)
        v2f xy = __builtin_nontemporal_load(&x[p]);
        const float px = xy.x;
        const float py = xy.y;

        float o[2 * NLV];

        #pragma unroll
        for (int l = 0; l < NLV; ++l) {
            const int   res = kRes[l];
            const float g   = 1024.0f / (float)res;   // IMG_W / res
            const float ig  = (float)res / 1024.0f;

            const int bx = (int)(px * ig);            // x >= 0 -> trunc == floor
            const int by = (int)(py * ig);
            const float wx = (px - (float)bx * g) * ig;
            const float wy = (py - (float)by * g) * ig;

            const int i00 = bx * res + by;            // reference's exact indexing

            v2f e00, e01, e10, e11;
            if (l < NCACHED) {
                const v2f* t = &lds_tab[kOff[l]];     // ds_load_b64 gathers
                e00 = t[i00];
                e01 = t[i00 + 1];
                e10 = t[i00 + res];
                e11 = t[i00 + res + 1];
            } else {
                const v2f* __restrict__ t = tabs.t[l]; // L2-resident gathers
                e00 = t[i00];
                e01 = t[i00 + 1];
                e10 = t[i00 + res];
                e11 = t[i00 + res + 1];
            }

            const v2f c0 = e00 * (1.0f - wy) + e01 * wy;
            const v2f c1 = e10 * (1.0f - wy) + e11 * wy;
            const v2f r  = c0 * (1.0f - wx) + c1 * wx;
            o[2 * l]     = r.x;
            o[2 * l + 1] = r.y;
        }

        // streaming output: 8x b128 non-temporal stores (row is 128B aligned)
        v4f* orow = (v4f*)(out + (size_t)p * (2 * NLV));
        #pragma unroll
        for (int k = 0; k < (2 * NLV) / 4; ++k) {
            v4f v = { o[4 * k], o[4 * k + 1], o[4 * k + 2], o[4 * k + 3] };
            __builtin_nontemporal_store(v, orow + k);
        }
    }
}

extern "C" void kernel_launch(void* const* d_in, const int* in_sizes, int n_in,
                              void* d_out, int out_size, void* d_ws, size_t ws_size,
                              hipStream_t stream) {
    (void)n_in; (void)d_ws; (void)ws_size; (void)out_size;

    const v2f* x = (const v2f*)d_in[0];
    Tabs tabs;
    for (int i = 0; i < NLV; ++i) tabs.t[i] = (const v2f*)d_in[1 + i];

    const int npts = in_sizes[0] / 2;
    int nblk = (npts + BLOCK - 1) / BLOCK;
    if (nblk > NBLOCKS) nblk = NBLOCKS;

    hashgrid_kernel<<<nblk, BLOCK, 0, stream>>>(x, tabs, (float*)d_out, npts);
}